// spe_self_atten_76398878261339
// MI455X (gfx1250) — compile-verified
//
#include <hip/hip_runtime.h>
#include <math.h>

typedef __attribute__((ext_vector_type(2))) float v2f;
typedef __attribute__((ext_vector_type(8))) float v8f;

#define BATCH 256
#define C 200
#define L 1024
#define CP 208          // padded c (13 * 16)
#define NT 13           // 16x16 tiles along c
#define NWAVES 13
#define NTHREADS (NWAVES * 32)
#define KC 32           // phase-1 k-chunk (columns of X staged per step)
#define XSS 36          // phase-1 LDS row stride (floats): 144B = 16-aligned, odd*4 banks ok
#define APS 210         // attention-prob LDS row stride (floats), even for b64
#define NC 64           // phase-2 column chunk (4 n-tiles)
#define EPSV 1e-8f

__global__ __launch_bounds__(NTHREADS)
void cos_attn_kernel(const float* __restrict__ x, float* __restrict__ out) {
    __shared__ float apm[CP * APS];   // softmax(A) probabilities, padded stride
    __shared__ float norms[CP];       // ||x_i||
    __shared__ union {
        float xs[2][CP * XSS];        // phase-1 double-buffered X k-chunks
        float xb[C * NC];             // phase-2 X column slab
    } u;

    const int b    = blockIdx.x;
    const int tid  = threadIdx.x;
    const int w    = tid >> 5;        // wave id 0..12 -> owns S rows 16w..16w+15
    const int lane = tid & 31;
    const int l16  = lane & 15;
    const int hi   = lane >> 4;       // half-wave select (K/M split per WMMA layout)

    const float* xg = x + (size_t)b * C * L;

    // ---------------- Phase 1: S = X * X^T (fp32 WMMA, K streamed) -------------
    v8f acc[NT] = {};                 // 13 accumulator tiles (row of tiles w)

    auto load_chunk = [&](int kb0, int buf) {
        // 208 rows x 32 cols = 1664 float4, 416 threads x 4
        #pragma unroll
        for (int i = 0; i < 4; ++i) {
            int f   = tid + i * NTHREADS;
            int row = f >> 3;         // 8 float4 per row
            int c4  = f & 7;
            float4 v = make_float4(0.f, 0.f, 0.f, 0.f);
            if (row < C)
                v = *(const float4*)(xg + (size_t)row * L + kb0 + c4 * 4);
            *(float4*)&u.xs[buf][row * XSS + c4 * 4] = v;   // 16B aligned (36*4=144)
        }
    };

    load_chunk(0, 0);
    __syncthreads();

    const int rbase = l16 * XSS + 2 * hi;     // fragment gather base (per lane)
    for (int kc = 0; kc < L / KC; ++kc) {
        int buf = kc & 1;
        if (kc + 1 < L / KC) load_chunk((kc + 1) * KC, buf ^ 1);
        const float* xs = u.xs[buf];
        #pragma unroll 2
        for (int ks = 0; ks < KC; ks += 4) {
            // A fragment for tile-row w (rows 16w+l16, k = ks + {0,1}/{2,3})
            v2f af = *(const v2f*)&xs[w * 16 * XSS + rbase + ks];
            v2f fr[NT];
            #pragma unroll
            for (int t = 0; t < NT; ++t)
                fr[t] = *(const v2f*)&xs[t * 16 * XSS + rbase + ks];
            #pragma unroll
            for (int t = 0; t < NT; ++t)
                acc[t] = __builtin_amdgcn_wmma_f32_16x16x4_f32(
                    false, af, false, fr[t], (short)0, acc[t], false, false);
        }
        __syncthreads();   // buf consumed; next iter may overwrite it
    }

    // ---- norms from the diagonal tile (w,w): S[i][i] = ||x_i||^2 ----
    #pragma unroll
    for (int t = 0; t < NT; ++t) {
        if (t == w) {
            #pragma unroll
            for (int g = 0; g < 8; ++g) {
                // element (M,N)=(g+8*hi, l16); diagonal when l16 == g+8*hi
                if (l16 == g + 8 * hi)
                    norms[16 * w + g + 8 * hi] = sqrtf(acc[t][g]);
            }
        }
    }
    __syncthreads();

    // ---- cosine scale + masked softmax, all in registers ----
    float ni[8];
    #pragma unroll
    for (int g = 0; g < 8; ++g) ni[g] = norms[16 * w + g + 8 * hi];

    float rmax[8];
    #pragma unroll
    for (int g = 0; g < 8; ++g) rmax[g] = -INFINITY;

    #pragma unroll
    for (int t = 0; t < NT; ++t) {
        float nj    = norms[16 * t + l16];
        bool  valid = (16 * t + l16) < C;       // mask padded columns
        #pragma unroll
        for (int g = 0; g < 8; ++g) {
            float s = valid ? acc[t][g] / fmaxf(ni[g] * nj, EPSV) : -INFINITY;
            acc[t][g] = s;
            rmax[g] = fmaxf(rmax[g], s);
        }
    }
    #pragma unroll
    for (int g = 0; g < 8; ++g) {               // reduce across the 16-lane half
        float m = rmax[g];
        #pragma unroll
        for (int off = 1; off <= 8; off <<= 1)
            m = fmaxf(m, __shfl_xor(m, off, 32));
        rmax[g] = m;
    }
    float rsum[8];
    #pragma unroll
    for (int g = 0; g < 8; ++g) rsum[g] = 0.f;
    #pragma unroll
    for (int t = 0; t < NT; ++t)
        #pragma unroll
        for (int g = 0; g < 8; ++g) {
            float e = __expf(acc[t][g] - rmax[g]);   // exp(-inf)=0 masks pads
            acc[t][g] = e;
            rsum[g] += e;
        }
    #pragma unroll
    for (int g = 0; g < 8; ++g) {
        float s = rsum[g];
        #pragma unroll
        for (int off = 1; off <= 8; off <<= 1)
            s += __shfl_xor(s, off, 32);
        rsum[g] = 1.f / s;
    }
    // store probabilities (rows 16w..16w+15) to LDS for phase-2 A fragments
    #pragma unroll
    for (int t = 0; t < NT; ++t)
        #pragma unroll
        for (int g = 0; g < 8; ++g)
            apm[(16 * w + g + 8 * hi) * APS + 16 * t + l16] = acc[t][g] * rsum[g];

    // ---------------- Phase 2: O = A * X + X (fp32 WMMA) ----------------------
    for (int nch = 0; nch < L / NC; ++nch) {
        __syncthreads();   // protect xb (aliases xs) vs previous readers
        #pragma unroll
        for (int i = 0; i < 8; ++i) {          // stage X[0..199][64 cols]
            int f = tid + i * NTHREADS;
            if (f < C * (NC / 4)) {
                int row = f / (NC / 4);
                int c4  = f % (NC / 4);
                *(float4*)&u.xb[row * NC + c4 * 4] =
                    *(const float4*)(xg + (size_t)row * L + nch * NC + c4 * 4);
            }
        }
        __syncthreads();

        v8f oacc[4] = {};
        for (int kb = 0; kb < C; kb += 4) {    // K = 200, exact (no pad reads)
            v2f af = *(const v2f*)&apm[(16 * w + l16) * APS + kb + 2 * hi];
            #pragma unroll
            for (int nt = 0; nt < 4; ++nt) {
                v2f bf;
                bf.x = u.xb[(kb + 2 * hi) * NC + nt * 16 + l16];
                bf.y = u.xb[(kb + 2 * hi + 1) * NC + nt * 16 + l16];
                oacc[nt] = __builtin_amdgcn_wmma_f32_16x16x4_f32(
                    false, af, false, bf, (short)0, oacc[nt], false, false);
            }
        }
        // residual add + coalesced store (rows >= 200 dropped)
        #pragma unroll
        for (int nt = 0; nt < 4; ++nt)
            #pragma unroll
            for (int g = 0; g < 8; ++g) {
                int row = 16 * w + g + 8 * hi;
                if (row < C) {
                    int col = nch * NC + nt * 16 + l16;
                    out[(size_t)b * C * L + (size_t)row * L + col] =
                        oacc[nt][g] + u.xb[row * NC + nt * 16 + l16];
                }
            }
    }
}

extern "C" void kernel_launch(void* const* d_in, const int* in_sizes, int n_in,
                              void* d_out, int out_size, void* d_ws, size_t ws_size,
                              hipStream_t stream) {
    const float* x = (const float*)d_in[0];
    float* out = (float*)d_out;
    cos_attn_kernel<<<dim3(BATCH), dim3(NTHREADS), 0, stream>>>(x, out);
}